// GraphPropagation_36842229465246
// MI455X (gfx1250) — compile-verified
//
#include <hip/hip_runtime.h>
#include <stdint.h>

// GraphPropagation on MI455X (gfx1250, wave32).
// Phase 1: pack neighbor matrix (0.0/1.0 floats) into bitmasks (256 MB -> 8 MB,
//          one streaming pass at HBM roofline; 8 MB stays resident in 192 MB L2).
// Phase 2: per-batch greedy loop (1 workgroup = 8 wave32s per batch).
//          (value,index) packed into one u64 key -> single-max reductions;
//          2 barriers/step; global-argmax fallback only on empty neighborhood;
//          wave-winner rows speculatively prefetched (global_prefetch_b8);
//          selected row pulled L2 -> LDS via CDNA5 async engine
//          (global_load_async_to_lds_b128 + s_wait_asynccnt).
// Phase 3: bandwidth-bound vertices[perm] row gather with B128 accesses.

#define BATCH 16
#define NV    2048
#define DIM   128
#define NCLS  8
#define TPB   256
#define KPT   (NV / TPB)   // 8 strided elements per thread

static __device__ __forceinline__ unsigned int ballot32(bool pred) {
#if __has_builtin(__builtin_amdgcn_ballot_w32)
    return __builtin_amdgcn_ballot_w32(pred);
#else
    return (unsigned int)__ballot(pred);
#endif
}

static __device__ __forceinline__ unsigned long long shfl_xor_u64(unsigned long long v, int m) {
    int lo = __shfl_xor((int)(unsigned int)v, m, 32);
    int hi = __shfl_xor((int)(unsigned int)(v >> 32), m, 32);
    return ((unsigned long long)(unsigned int)hi << 32) | (unsigned int)lo;
}

// key = (float_bits(value) << 32) | ~index : conf >= 0 so u64 max ==
// (max value, then min index) == jnp.argmax first-max tie-break. key==0 <=> no entry.
static __device__ __forceinline__ unsigned long long mkkey(float v, int i) {
    return ((unsigned long long)__float_as_uint(v) << 32) | (unsigned int)(~i);
}

// ---------------------------------------------------------------- phase 1
__global__ __launch_bounds__(TPB) void pack_neighbors_kernel(
        const float* __restrict__ nbr, unsigned int* __restrict__ bits) {
    const int r    = blockIdx.x;          // row in [0, BATCH*NV)
    const int t    = threadIdx.x;
    const int lane = t & 31;
    const size_t base = (size_t)r * NV;
    float v[KPT];
#pragma unroll
    for (int k = 0; k < KPT; ++k)         // 8 loads in flight per lane
        v[k] = nbr[base + (k << 8) + t];
#pragma unroll
    for (int k = 0; k < KPT; ++k) {
        const unsigned int m = ballot32(v[k] == 1.0f);
        if (lane == 0) bits[(size_t)r * (NV / 32) + (((k << 8) + t) >> 5)] = m;
    }
}

// ---------------------------------------------------------------- phase 2
__global__ __launch_bounds__(TPB) void greedy_kernel(
        const float* __restrict__ confidence,   // [BATCH, NV, NCLS]
        const float* __restrict__ neighbors,    // [BATCH, NV, NV]
        const unsigned int* __restrict__ bits,  // [BATCH*NV, NV/32] (optional)
        int* __restrict__ perm_out,             // [BATCH, NV]
        int use_bits) {
    __shared__ unsigned int       nrow[NV / 32];   // 256 B: selected row bits
    __shared__ unsigned long long wkey[TPB / 32];  // per-wave reduction keys

    const int b    = blockIdx.x;
    const int t    = threadIdx.x;
    const int w    = t >> 5;
    const int lane = t & 31;

    // conf0 = max over classes; this thread's 8 strided values live in registers.
    float cr[KPT];
#pragma unroll
    for (int k = 0; k < KPT; ++k) {
        const int i = t + (k << 8);
        const float4* cp = (const float4*)(confidence + ((size_t)b * NV + i) * NCLS);
        const float4 a = cp[0], c = cp[1];
        cr[k] = fmaxf(fmaxf(fmaxf(a.x, a.y), fmaxf(a.z, a.w)),
                      fmaxf(fmaxf(c.x, c.y), fmaxf(c.z, c.w)));
    }
    unsigned int nb = 0u;                       // neighborhood bits (per-thread)
    unsigned int uv = (1u << KPT) - 1u;         // unvisited bits (per-thread)

    for (int step = 0; step < NV; ++step) {
        // ---- masked (neighborhood) argmax as u64-key max
        unsigned long long mkey = 0ull;
#pragma unroll
        for (int k = 0; k < KPT; ++k) {
            if ((nb >> k) & 1u) {
                const unsigned long long kk = mkkey(cr[k], t + (k << 8));
                if (kk > mkey) mkey = kk;
            }
        }
#pragma unroll
        for (int off = 16; off > 0; off >>= 1) {
            const unsigned long long o = shfl_xor_u64(mkey, off);
            if (o > mkey) mkey = o;
        }
        if (lane == 0) {
            wkey[w] = mkey;
            if (use_bits && mkey != 0ull) {
                // speculative: winner is one of the 8 wave winners; start moving
                // its 256 B bitmask row L2 -> WGP$ now (global_prefetch_b8).
                const int fw = (int)~(unsigned int)mkey;
                const char* p = (const char*)(bits + ((size_t)b * NV + fw) * (NV / 32));
                __builtin_prefetch(p, 0, 3);
                __builtin_prefetch(p + 128, 0, 3);
            }
        }
        __syncthreads();                                   // barrier A

        // ---- all threads redundantly reduce the 8 wave keys (uniform result)
        unsigned long long key = wkey[0];
#pragma unroll
        for (int q = 1; q < TPB / 32; ++q)
            if (wkey[q] > key) key = wkey[q];

        if (key == 0ull) {
            // ---- rare: neighborhood empty -> global argmax over conf
            __syncthreads();   // protect wkey reuse across skewed waves
            unsigned long long gkey = 0ull;
#pragma unroll
            for (int k = 0; k < KPT; ++k) {
                const unsigned long long kk = mkkey(cr[k], t + (k << 8));
                if (kk > gkey) gkey = kk;
            }
#pragma unroll
            for (int off = 16; off > 0; off >>= 1) {
                const unsigned long long o = shfl_xor_u64(gkey, off);
                if (o > gkey) gkey = o;
            }
            if (lane == 0) wkey[w] = gkey;
            __syncthreads();
            key = wkey[0];
#pragma unroll
            for (int q = 1; q < TPB / 32; ++q)
                if (wkey[q] > key) key = wkey[q];
        }

        const int   fc  = (int)~(unsigned int)key;             // winning index
        const float cfc = __uint_as_float((unsigned int)(key >> 32)); // == conf[fc]
        if (t == 0) perm_out[(size_t)b * NV + step] = fc;

        // ---- fetch neighbor row fc (bits) into LDS
        if (use_bits) {
            if (t < 16) {   // 16 lanes x B128 = 256 B row, async L2/WGP$ -> LDS
                const unsigned int loff = (unsigned int)(uintptr_t)(&nrow[t * 4]);
                const unsigned long long ga = (unsigned long long)(uintptr_t)(
                    bits + ((size_t)b * NV + fc) * (NV / 32) + t * 4);
                asm volatile("global_load_async_to_lds_b128 %0, %1, off"
                             :: "v"(loff), "v"(ga) : "memory");
            }
        } else {
            const float* rowp = neighbors + ((size_t)b * NV + fc) * NV;
            float rv[KPT];
#pragma unroll
            for (int k = 0; k < KPT; ++k) rv[k] = rowp[(k << 8) + t];
#pragma unroll
            for (int k = 0; k < KPT; ++k) {
                const unsigned int m = ballot32(rv[k] == 1.0f);
                if (lane == 0) nrow[(((k << 8) + t) >> 5)] = m;
            }
        }

        // ---- conf = conf * (conf != conf[fc])   (overlaps with async fetch)
#pragma unroll
        for (int k = 0; k < KPT; ++k)
            if (cr[k] == cfc) cr[k] = 0.0f;
        // ---- clear fc from this thread's masks (neighbors diag is zero, so
        //      clearing unvis before the OR matches the reference exactly)
        {
            const int rel = fc - t;
            if (rel >= 0 && (rel & 255) == 0) {
                const unsigned int bit = 1u << (rel >> 8);
                nb &= ~bit; uv &= ~bit;
            }
        }

        asm volatile("s_wait_asynccnt 0" ::: "memory");
        __syncthreads();                                   // barrier B

        // ---- nbhd |= neighbors[fc] & unvisited
#pragma unroll
        for (int k = 0; k < KPT; ++k) {
            const unsigned int word = nrow[(t >> 5) + 8 * k]; // LDS broadcast read
            nb |= (((word >> (t & 31)) & (uv >> k)) & 1u) << k;
        }
    }
}

// ---------------------------------------------------------------- phase 3
__global__ __launch_bounds__(TPB) void gather_rows_kernel(
        const float* __restrict__ verts, const int* __restrict__ perm,
        float* __restrict__ outv) {
    const int t    = threadIdx.x;
    const int r    = blockIdx.x * (TPB / 32) + (t >> 5);  // row in [0, BATCH*NV)
    const int lane = t & 31;
    const int b    = r >> 11;                             // r / NV
    const int p    = perm[r];
    const float4* src = (const float4*)(verts + ((size_t)b * NV + p) * DIM);
    float4*       dst = (float4*)(outv + (size_t)r * DIM);
    dst[lane] = src[lane];                                // 32 lanes x float4 = 128 f32
}

// ---------------------------------------------------------------- launch
extern "C" void kernel_launch(void* const* d_in, const int* in_sizes, int n_in,
                              void* d_out, int out_size, void* d_ws, size_t ws_size,
                              hipStream_t stream) {
    (void)in_sizes; (void)n_in; (void)out_size;

    const float* verts = (const float*)d_in[0];   // [B, N, D]  f32
    const float* confd = (const float*)d_in[1];   // [B, N, C]  f32
    const float* nbrs  = (const float*)d_in[2];   // [B, N, N]  f32

    float* outv = (float*)d_out;                               // [B, N, D] f32
    int*   outp = (int*)(outv + (size_t)BATCH * NV * DIM);     // [B, N]    i32

    unsigned int* bits = (unsigned int*)d_ws;
    const size_t bits_bytes =
        (size_t)BATCH * NV * (NV / 32) * sizeof(unsigned int); // 8 MB
    const int use_bits = (d_ws != nullptr && ws_size >= bits_bytes) ? 1 : 0;

    if (use_bits) {
        pack_neighbors_kernel<<<BATCH * NV, TPB, 0, stream>>>(nbrs, bits);
    }
    greedy_kernel<<<BATCH, TPB, 0, stream>>>(confd, nbrs, bits, outp, use_bits);
    gather_rows_kernel<<<(BATCH * NV) / (TPB / 32), TPB, 0, stream>>>(verts, outp, outv);
}